// ContrastiveLoss_56977036148935
// MI455X (gfx1250) — compile-verified
//
#include <hip/hip_runtime.h>
#include <hip/hip_bf16.h>

typedef float v2f __attribute__((ext_vector_type(2)));
typedef float v8f __attribute__((ext_vector_type(8)));

#define CN 8192
#define CD 64
#define CMARGIN 1.0f

// ---------------------------------------------------------------------------
// Kernel 0: per-row squared norms  sq[i] = sum_k out[i][k]^2
// ---------------------------------------------------------------------------
__global__ __launch_bounds__(256) void cl_sq_kernel(const float* __restrict__ out,
                                                    float* __restrict__ sq) {
    int i = blockIdx.x * 256 + threadIdx.x;
    if (i < CN) {
        const float4* row = (const float4*)(out + (size_t)i * CD);
        float s = 0.0f;
#pragma unroll
        for (int q = 0; q < CD / 4; ++q) {
            float4 v = row[q];
            s += v.x * v.x + v.y * v.y + v.z * v.z + v.w * v.w;
        }
        sq[i] = s;
    }
}

// ---------------------------------------------------------------------------
// Kernel 1: fused Gram-tile (WMMA f32 16x16x4) + contrastive-loss epilogue.
// Each wave owns one 16x16 tile of the 8192x8192 pair matrix.
// Upper triangle only: off-diagonal tiles scaled by 2, diagonal tiles mask i==j.
// Block = 256 threads = 8 waves covering a 64(rows) x 32(cols) super-tile.
// Grid  = (8192/64, 8192/32) = (128, 256).
// One deterministic partial per block (no float atomics).
// ---------------------------------------------------------------------------
__global__ __launch_bounds__(256) void cl_tile_kernel(const float* __restrict__ outm,
                                                      const int* __restrict__ label,
                                                      const float* __restrict__ sq,
                                                      float* __restrict__ partials) {
    __shared__ float red[8];

    const int wave = threadIdx.x >> 5;
    const int lane = threadIdx.x & 31;
    const int wi = wave >> 1;          // 0..3  (row sub-tile)
    const int wj = wave & 1;           // 0..1  (col sub-tile)
    const int i0 = blockIdx.x * 64 + wi * 16;
    const int j0 = blockIdx.y * 32 + wj * 16;

    float acc = 0.0f;

    if (j0 >= i0) {                    // wave-uniform: EXEC stays all-ones for WMMA
        const float scale = (j0 == i0) ? 1.0f : 2.0f;

        // Fragment addressing (wave32 WMMA f32 16x16x4 layout):
        //  A (16x4, MxK): lane l, VGPR v -> A[M = l&15][K = 2*(l>>4) + v]
        //  B ( 4x16, KxN): lane l, VGPR v -> B[K = 2*(l>>4) + v][N = l&15]
        // With B = (row block j0)^T, both operands load element
        //  mat[base + (l&15)][4*s + 2*(l>>4) + {0,1}]  -> one b64 load each.
        const int m0 = lane & 15;
        const int kh = lane >> 4;
        const float* arow = outm + (size_t)(i0 + m0) * CD + 2 * kh;
        const float* brow = outm + (size_t)(j0 + m0) * CD + 2 * kh;

        v8f c = {0.f, 0.f, 0.f, 0.f, 0.f, 0.f, 0.f, 0.f};
#pragma unroll
        for (int s = 0; s < CD / 4; ++s) {
            v2f a = *(const v2f*)(arow + 4 * s);
            v2f b = *(const v2f*)(brow + 4 * s);
            c = __builtin_amdgcn_wmma_f32_16x16x4_f32(
                    /*neg_a=*/false, a, /*neg_b=*/false, b,
                    /*c_mod=*/(short)0, c, /*reuse_a=*/false, /*reuse_b=*/false);
        }

        // Epilogue: C/D layout -> lane l, VGPR r holds (M = r + 8*(l>>4), N = l&15)
        const int n    = lane & 15;
        const int j    = j0 + n;
        const float sqj = sq[j];
        const int   labj = label[j];
#pragma unroll
        for (int r = 0; r < 8; ++r) {
            const int i = i0 + r + 8 * kh;
            float d2 = fmaxf(sq[i] + sqj - 2.0f * c[r], 0.0f);
            float term;
            if (label[i] == labj) {
                term = d2;                               // positive pair: d^2
            } else {
                float d = sqrtf(d2);
                float t = fmaxf(CMARGIN - d, 0.0f);      // negative pair: hinge^2
                term = t * t;
            }
            if (i != j) acc += term;                     // mask diagonal
        }
        acc *= scale;
    }

    // wave32 shuffle reduction
#pragma unroll
    for (int off = 16; off > 0; off >>= 1) acc += __shfl_down(acc, off, 32);
    if (lane == 0) red[wave] = acc;
    __syncthreads();

    if (threadIdx.x == 0) {
        float s = 0.0f;
#pragma unroll
        for (int w = 0; w < 8; ++w) s += red[w];
        partials[blockIdx.x * gridDim.y + blockIdx.y] = s;
    }
}

// ---------------------------------------------------------------------------
// Kernel 2: deterministic final reduction of per-block partials.
// ---------------------------------------------------------------------------
__global__ __launch_bounds__(256) void cl_reduce_kernel(const float* __restrict__ partials,
                                                        int count,
                                                        float* __restrict__ outv) {
    __shared__ float red[256];
    float s = 0.0f;
    for (int idx = threadIdx.x; idx < count; idx += 256) s += partials[idx];
    red[threadIdx.x] = s;
    __syncthreads();
#pragma unroll
    for (int stride = 128; stride > 0; stride >>= 1) {
        if (threadIdx.x < stride) red[threadIdx.x] += red[threadIdx.x + stride];
        __syncthreads();
    }
    if (threadIdx.x == 0)
        outv[0] = red[0] * (1.0f / ((float)CN * (float)(CN - 1)));
}

extern "C" void kernel_launch(void* const* d_in, const int* in_sizes, int n_in,
                              void* d_out, int out_size, void* d_ws, size_t ws_size,
                              hipStream_t stream) {
    const float* outm  = (const float*)d_in[0];   // [8192, 64] fp32
    const int*   label = (const int*)d_in[1];     // [8192] int
    float*       loss  = (float*)d_out;           // scalar

    float* sq       = (float*)d_ws;               // 8192 floats
    float* partials = sq + CN;                    // 128*256 = 32768 floats

    // 1) row squared norms
    cl_sq_kernel<<<dim3(CN / 256), dim3(256), 0, stream>>>(outm, sq);

    // 2) fused WMMA pair-loss tiles
    dim3 grid(CN / 64, CN / 32);                  // (128, 256)
    cl_tile_kernel<<<grid, dim3(256), 0, stream>>>(outm, label, sq, partials);

    // 3) final reduction
    cl_reduce_kernel<<<dim3(1), dim3(256), 0, stream>>>(partials, grid.x * grid.y, loss);
}